// FPSModel_80753975099708
// MI455X (gfx1250) — compile-verified
//
#include <hip/hip_runtime.h>

// ---------------------------------------------------------------------------
// Farthest Point Sampling, MI455X (gfx1250, wave32).
//   x: [B, 32768, 3] f32  ->  d_out = [B*1024*3 f32 sampled][B*1024 i32 idx]
//
// One workgroup (1024 threads / 32 waves) per batch; 1024 inherently
// sequential argmax rounds. Point coords live in VGPRs (32 pts/lane, fits the
// 128-VGPR / 8-waves-per-SIMD budget with no spills), min_d lives in LDS
// (128 KB float4 layout -> conflict-free ds_load_b128/ds_store_b128).
// Wave argmax: ds_swizzle xor-butterflies. Cross-wave argmax: one packed-key
// ds_max_u64 atomic per wave leader (bit pattern of a non-negative f32 is
// order-isomorphic; low word = ~gid gives first-index tie-break), with a
// parity ping-pong so the key slot is re-zeroed without an extra barrier.
// Only 2 workgroup barriers per round.
//
// WMMA is intentionally unused: FPS has no matmul-shaped work. A 16x16x4 f32
// WMMA yields 128 useful FLOP per 2048 and its D-tile extraction into
// per-point min/argmax state costs more VALU than the FMAs it replaces.
// ---------------------------------------------------------------------------

constexpr int kN      = 32768;  // points per cloud
constexpr int kS      = 1024;   // samples
constexpr int kTPB    = 1024;   // threads per block (32 waves)
constexpr int kChunks = 8;      // 8 chunks * 4 points = 32 points / thread

// wave32 xor-butterfly lane exchange via DS_SWIZZLE_B32 (group-of-32 mode:
// offset = {1'b0, xor[4:0], or[4:0]=0, and[4:0]=0x1f}).
template <int MASK>
__device__ __forceinline__ int swz_i(int v) {
  return __builtin_amdgcn_ds_swizzle(v, (MASK << 10) | 0x1f);
}
template <int MASK>
__device__ __forceinline__ float swz_f(float v) {
  return __int_as_float(swz_i<MASK>(__float_as_int(v)));
}

template <int MASK>
__device__ __forceinline__ void bfly(float& v, int& i) {
  float ov = swz_f<MASK>(v);
  int   oi = swz_i<MASK>(i);
  // argmax combine with first-index tie-break (matches jnp.argmax)
  bool take = (ov > v) || ((ov == v) && (oi < i));
  v = take ? ov : v;
  i = take ? oi : i;
}

__global__ __launch_bounds__(kTPB, 1) void fps_kernel(
    const float* __restrict__ x,
    float* __restrict__ out_pts,
    int* __restrict__ out_idx) {
  const int b    = blockIdx.x;
  const int t    = threadIdx.x;
  const int lane = t & 31;

  __shared__ float s_px, s_py, s_pz;            // coords of current farthest pt
  __shared__ int   s_cur;                       // its global index
  __shared__ unsigned long long s_key[2];       // ping-pong packed argmax keys
  extern __shared__ float4 md4[];               // min_d: [kChunks][kTPB] = 128 KB

  // ---- load this thread's 32 points into registers (coalesced b128) -------
  // Ownership: global point g = (c << 12) | (t << 2) | j,  c in [0,8), j in [0,4)
  float px[32], py[32], pz[32];
  const float4* xb = (const float4*)(x + (size_t)b * kN * 3);
  const float inf = __builtin_inff();
#pragma unroll
  for (int c = 0; c < kChunks; ++c) {
    const int base = c * 3072 + t * 3;  // float4 index (48 B / thread / chunk)
    float4 a0 = xb[base + 0];
    float4 a1 = xb[base + 1];
    float4 a2 = xb[base + 2];
    px[c * 4 + 0] = a0.x; py[c * 4 + 0] = a0.y; pz[c * 4 + 0] = a0.z;
    px[c * 4 + 1] = a0.w; py[c * 4 + 1] = a1.x; pz[c * 4 + 1] = a1.y;
    px[c * 4 + 2] = a1.z; py[c * 4 + 2] = a1.w; pz[c * 4 + 2] = a2.x;
    px[c * 4 + 3] = a2.y; py[c * 4 + 3] = a2.z; pz[c * 4 + 3] = a2.w;
    md4[c * kTPB + t] = make_float4(inf, inf, inf, inf);
  }
  if (t == 0) {  // deterministic seed: point 0 (thread 0, chunk 0, j 0)
    s_cur = 0;
    s_px = px[0]; s_py = py[0]; s_pz = pz[0];
    s_key[0] = 0ull; s_key[1] = 0ull;
  }
  __syncthreads();

  float* outp = out_pts + (size_t)b * kS * 3;
  int*   outi = out_idx + (size_t)b * kS;

#pragma unroll 1
  for (int s = 0; s < kS; ++s) {
    const int   kb = s & 1;
    const float sx = s_px, sy = s_py, sz = s_pz;
    const int   cur = s_cur;
    if (t == 0) {  // emit the index consumed this round + its exact coords
      outp[s * 3 + 0] = sx;
      outp[s * 3 + 1] = sy;
      outp[s * 3 + 2] = sz;
      outi[s] = cur;
      s_key[1 - kb] = 0ull;  // pre-zero next round's key (race-free: next
                             // round's atomics start after this round's barrier)
    }

    // ---- distance update + local argmax over this lane's 32 points --------
    float best = -1.0f;  // all min_d >= 0
    int   bidx = 0;
#pragma unroll
    for (int c = 0; c < kChunks; ++c) {
      float4 m = md4[c * kTPB + t];  // ds_load_b128, conflict-free
#define FPS_POINT(J, FLD)                                                      \
      {                                                                        \
        float dx = px[c * 4 + J] - sx;                                         \
        float dy = py[c * 4 + J] - sy;                                         \
        float dz = pz[c * 4 + J] - sz;                                         \
        float d  = fmaf(dz, dz, fmaf(dy, dy, dx * dx));                        \
        m.FLD = fminf(m.FLD, d);                                               \
        bool p = m.FLD > best; /* strict > keeps first index (ascending g) */  \
        best = p ? m.FLD : best;                                               \
        bidx = p ? ((c << 12) | (t << 2) | J) : bidx;                          \
      }
      FPS_POINT(0, x)
      FPS_POINT(1, y)
      FPS_POINT(2, z)
      FPS_POINT(3, w)
#undef FPS_POINT
      md4[c * kTPB + t] = m;  // ds_store_b128
    }

    // ---- wave32 xor-butterfly argmax (ds_swizzle) -------------------------
    bfly<16>(best, bidx);
    bfly<8>(best, bidx);
    bfly<4>(best, bidx);
    bfly<2>(best, bidx);
    bfly<1>(best, bidx);

    // ---- cross-wave argmax: one packed-key ds_max_u64 per wave leader -----
    // key = {f32 bits of best (monotone for best >= 0), ~gid (first-idx ties)}
    if (lane == 0) {
      unsigned long long key =
          ((unsigned long long)(unsigned)__float_as_int(best) << 32) |
          (unsigned long long)(unsigned)(~bidx);
      atomicMax(&s_key[kb], key);
    }
    __syncthreads();

    // ---- decode winner; its owner publishes coords from registers ---------
    const int g = (int)(~(unsigned)(s_key[kb] & 0xffffffffull));
    if (((g >> 2) & (kTPB - 1)) == t) {
      float bx = 0.f, by = 0.f, bz = 0.f;
#pragma unroll
      for (int c = 0; c < kChunks; ++c) {
#pragma unroll
        for (int j = 0; j < 4; ++j) {
          const int gid = (c << 12) | (t << 2) | j;
          bool m = (gid == g);
          bx = m ? px[c * 4 + j] : bx;
          by = m ? py[c * 4 + j] : by;
          bz = m ? pz[c * 4 + j] : bz;
        }
      }
      s_px = bx; s_py = by; s_pz = bz;
    }
    if (t == 0) s_cur = g;
    __syncthreads();
  }
}

extern "C" void kernel_launch(void* const* d_in, const int* in_sizes, int n_in,
                              void* d_out, int out_size, void* d_ws, size_t ws_size,
                              hipStream_t stream) {
  (void)n_in; (void)d_ws; (void)ws_size; (void)out_size;
  const float* x = (const float*)d_in[0];
  const int B = in_sizes[0] / (kN * 3);  // 64

  float* out_pts = (float*)d_out;                               // [B,1024,3] f32
  int*   out_idx = (int*)((float*)d_out + (size_t)B * kS * 3);  // [B,1024] i32

  const size_t shmem = (size_t)kChunks * kTPB * sizeof(float4);  // 128 KB
  (void)hipFuncSetAttribute((const void*)fps_kernel,
                            hipFuncAttributeMaxDynamicSharedMemorySize,
                            (int)shmem);
  hipLaunchKernelGGL(fps_kernel, dim3(B), dim3(kTPB), shmem, stream,
                     x, out_pts, out_idx);
}